// TopKPooling_25125558681608
// MI455X (gfx1250) — compile-verified
//
#include <hip/hip_runtime.h>

typedef __attribute__((ext_vector_type(2))) float v2f;
typedef __attribute__((ext_vector_type(4))) float v4f;
typedef __attribute__((ext_vector_type(8))) float v8f;

#define B_GRAPHS 64
#define N_PER    2048
#define K_SEL    1024
#define C_DIM    256
#define N_NODES  (B_GRAPHS * N_PER)   // 131072
#define N_EDGES  2097152

// ---------------------------------------------------------------------------
// Kernel 1: per-node score via V_WMMA_F32_16X16X4_F32.
// Each wave32 handles 16 nodes. A = weight chunk broadcast over all 16 rows
// (A 16x4 layout: VGPR0 holds K=0 / K=2 in lane halves, VGPR1 K=1 / K=3),
// B = 16 node rows as columns. Accumulating 64 chunks makes every D row equal
// to the 16 node scores; lane L element 0 = score(node L&15).
// ---------------------------------------------------------------------------
__global__ __launch_bounds__(256)
void score_kernel(const float* __restrict__ x, const float* __restrict__ weight,
                  float* __restrict__ score_ws, int* __restrict__ new_ids_ws) {
  __shared__ float s_w[C_DIM];
  __shared__ float s_red[256];
  const int tid = threadIdx.x;

  float w = weight[tid];          // C_DIM == blockDim.x == 256
  s_w[tid] = w;
  s_red[tid] = w * w;
  __syncthreads();
  for (int off = 128; off > 0; off >>= 1) {
    if (tid < off) s_red[tid] += s_red[tid + off];
    __syncthreads();
  }
  const float inv_norm = rsqrtf(s_red[0]);

  const int wave = tid >> 5;
  const int lane = tid & 31;
  const int m    = lane & 15;     // node within the wave's 16-node tile
  const int h    = lane >> 4;     // lane-half selects K pair {0,1} vs {2,3}
  const int node_base = blockIdx.x * 128 + wave * 16;
  const float* xrow = x + (size_t)(node_base + m) * C_DIM + 2 * h;

  v8f acc = {};
  for (int c = 0; c < 64; ++c) {
    v2f a, b;
    a[0] = s_w[4 * c + 2 * h];
    a[1] = s_w[4 * c + 2 * h + 1];
    b = *(const v2f*)(xrow + 4 * c);
    acc = __builtin_amdgcn_wmma_f32_16x16x4_f32(
        /*neg_a=*/false, a, /*neg_b=*/false, b,
        /*c_mod=*/(short)0, acc, /*reuse_a=*/false, /*reuse_b=*/false);
  }

  const float sc = tanhf(acc[0] * inv_norm);
  if (h == 0) {
    const int node = node_base + m;
    score_ws[node]   = sc;
    new_ids_ws[node] = -1;        // default: not selected
  }
}

// ---------------------------------------------------------------------------
// Kernel 2: per-graph top-k via in-LDS bitonic sort of (score, idx),
// descending by score, ascending index on ties (matches stable top_k).
// One 256-thread workgroup per graph; 2048 elements, 16 KB LDS.
// ---------------------------------------------------------------------------
__global__ __launch_bounds__(256)
void topk_kernel(const float* __restrict__ score_ws,
                 int* __restrict__ new_ids_ws, int* __restrict__ sel_nodes_ws) {
  __shared__ float s_key[N_PER];
  __shared__ int   s_idx[N_PER];
  const int tid  = threadIdx.x;
  const int b    = blockIdx.x;
  const int base = b * N_PER;

  for (int t = tid; t < N_PER; t += 256) {
    s_key[t] = score_ws[base + t];
    s_idx[t] = t;
  }
  __syncthreads();

  for (int k = 2; k <= N_PER; k <<= 1) {
    for (int j = k >> 1; j > 0; j >>= 1) {
      for (int t = tid; t < N_PER; t += 256) {
        const int ixj = t ^ j;
        if (ixj > t) {
          const float ki = s_key[t], kj = s_key[ixj];
          const int   vi = s_idx[t], vj = s_idx[ixj];
          // "t should precede ixj" under (score desc, idx asc)
          const bool cb = (ki > kj) || (ki == kj && vi < vj);
          const bool sw = ((t & k) == 0) ? (!cb) : cb;
          if (sw) {
            s_key[t] = kj; s_key[ixj] = ki;
            s_idx[t] = vj; s_idx[ixj] = vi;
          }
        }
      }
      __syncthreads();
    }
  }

  for (int r = tid; r < K_SEL; r += 256) {
    const int node = base + s_idx[r];
    const int pos  = b * K_SEL + r;       // output row / remapped node id
    new_ids_ws[node]  = pos;
    sel_nodes_ws[pos] = node;
  }
}

// ---------------------------------------------------------------------------
// Kernel 3: gather selected rows, gate by score, emit batch_out.
// x rows were just streamed by kernel 1 and (134 MB < 192 MB L2) are still
// L2-resident, so this pass is mostly an L2->HBM writeback.
// 4 rows per 256-thread block, float4 lanes.
// ---------------------------------------------------------------------------
__global__ __launch_bounds__(256)
void gather_kernel(const float* __restrict__ x, const float* __restrict__ score_ws,
                   const int* __restrict__ sel_nodes_ws,
                   float* __restrict__ x_out, float* __restrict__ batch_out) {
  const int tid  = threadIdx.x;
  const int row  = blockIdx.x * 4 + (tid >> 6);
  const int q    = tid & 63;
  const int node = sel_nodes_ws[row];
  const float s  = score_ws[node];

  v4f v = ((const v4f*)(x + (size_t)node * C_DIM))[q];
  v4f o = { v[0] * s, v[1] * s, v[2] * s, v[3] * s };
  ((v4f*)(x_out + (size_t)row * C_DIM))[q] = o;

  if (q == 0) batch_out[row] = (float)(row >> 10);   // row / K_SEL = graph id
}

// ---------------------------------------------------------------------------
// Kernel 4: edge filter + remap. Pure streaming; 1 edge per thread.
// ---------------------------------------------------------------------------
__global__ __launch_bounds__(256)
void edge_kernel(const int* __restrict__ edge_index, const int* __restrict__ new_ids_ws,
                 float* __restrict__ new_edge_out, float* __restrict__ mask_out) {
  const int e  = blockIdx.x * 256 + threadIdx.x;
  const int sN = edge_index[e];
  const int dN = edge_index[N_EDGES + e];
  const int ns = new_ids_ws[sN];
  const int nd = new_ids_ws[dN];
  const bool m = (ns >= 0) && (nd >= 0);
  new_edge_out[e]           = m ? (float)ns : -1.0f;
  new_edge_out[N_EDGES + e] = m ? (float)nd : -1.0f;
  mask_out[e]               = m ? 1.0f : 0.0f;
}

// ---------------------------------------------------------------------------
extern "C" void kernel_launch(void* const* d_in, const int* in_sizes, int n_in,
                              void* d_out, int out_size, void* d_ws, size_t ws_size,
                              hipStream_t stream) {
  (void)in_sizes; (void)n_in; (void)out_size; (void)ws_size;

  const float* x          = (const float*)d_in[0];
  // d_in[1] (x_score) is unused by the reference.
  const int*   edge_index = (const int*)d_in[2];
  // d_in[3] (batch) is implied by the uniform layout.
  const float* weight     = (const float*)d_in[4];

  // Outputs, concatenated flat in return order (all f32):
  float* out       = (float*)d_out;
  float* x_out     = out;                                          // 16,777,216
  float* edge_out  = x_out + (size_t)B_GRAPHS * K_SEL * C_DIM;     //  4,194,304
  float* mask_out  = edge_out + 2 * (size_t)N_EDGES;               //  2,097,152
  float* batch_out = mask_out + (size_t)N_EDGES;                   //     65,536

  // Workspace: scores | new_ids | selected node list  (~1.3 MB total)
  float* score_ws     = (float*)d_ws;
  int*   new_ids_ws   = (int*)(score_ws + N_NODES);
  int*   sel_nodes_ws = new_ids_ws + N_NODES;

  score_kernel <<<N_NODES / 128,           256, 0, stream>>>(x, weight, score_ws, new_ids_ws);
  topk_kernel  <<<B_GRAPHS,                256, 0, stream>>>(score_ws, new_ids_ws, sel_nodes_ws);
  gather_kernel<<<(B_GRAPHS * K_SEL) / 4,  256, 0, stream>>>(x, score_ws, sel_nodes_ws, x_out, batch_out);
  edge_kernel  <<<N_EDGES / 256,           256, 0, stream>>>(edge_index, new_ids_ws, edge_out, mask_out);
}